// ReachabilityGNNv13_61933428408480
// MI455X (gfx1250) — compile-verified
//
#include <hip/hip_runtime.h>
#include <hip/hip_bf16.h>
#include <math.h>

#define NN 100000
#define NE 1200000
#define ET (NE + NN)          // edges + self loops
#define NG 1000
#define HID 64
#define OUTC 4
#define ALPHA_FLAG 0.03f
#define NEG_SLOPE 0.2f
#define WT_LD 66              // padded LDS row stride (keeps b64 align, avoids bank conflicts)

typedef __attribute__((ext_vector_type(2))) float v2f;
typedef __attribute__((ext_vector_type(4))) float v4f;
typedef __attribute__((ext_vector_type(8))) float v8f;

// ---------------------------------------------------------------------------
// fp32 WMMA GEMM:  D[M x 64] = op( A[M x 64] @ W[64 x 64] (+ bias) )
// One wave -> 16 rows x 64 cols strip; 4 waves per block -> 64 rows.
// W staged transposed in LDS; A loaded as b64 with clamped rows (no exec
// masking inside the WMMA loop). V_WMMA_F32_16X16X4_F32, K in chunks of 4.
// ---------------------------------------------------------------------------
__global__ __launch_bounds__(128) void gat_gemm64_wmma(
    const float* __restrict__ A, const float* __restrict__ W,
    const float* __restrict__ bias, float* __restrict__ D,
    int M, int applyRelu)
{
    __shared__ float wt[HID * WT_LD];

    // stage W transposed: wt[n * WT_LD + k] = W[k * 64 + n]
    for (int i = threadIdx.x; i < HID * HID; i += 128) {
        int k = i >> 6, n = i & 63;
        wt[n * WT_LD + k] = W[i];
    }
    __syncthreads();

    const int lane = threadIdx.x & 31;
    const int wave = threadIdx.x >> 5;
    const int rowBase = (blockIdx.x * 4 + wave) * 16;
    const int m   = lane & 15;           // M (A) or N (B/C) index within tile
    const int kh  = (lane >> 4) << 1;    // K offset: 0 (lanes 0-15) or 2 (16-31)
    const int hi8 = (lane >> 4) << 3;    // C/D row offset: 0 or 8

    // clamp instead of predicate: garbage rows only feed garbage D rows,
    // which are masked at the store below.
    int r = rowBase + m;
    if (r >= M) r = M - 1;
    const float* arow = A + (size_t)r * HID;

    v8f acc0 = {}, acc1 = {}, acc2 = {}, acc3 = {};

    for (int k0 = 0; k0 < HID; k0 += 4) {
        v2f a = *(const v2f*)(arow + k0 + kh);            // global_load_b64
        const float* wp = wt + (k0 + kh);
        v2f b0 = *(const v2f*)(wp + (m     ) * WT_LD);    // ds_load_b64
        v2f b1 = *(const v2f*)(wp + (m + 16) * WT_LD);
        v2f b2 = *(const v2f*)(wp + (m + 32) * WT_LD);
        v2f b3 = *(const v2f*)(wp + (m + 48) * WT_LD);
        acc0 = __builtin_amdgcn_wmma_f32_16x16x4_f32(false, a, false, b0, (short)0, acc0, false, false);
        acc1 = __builtin_amdgcn_wmma_f32_16x16x4_f32(false, a, false, b1, (short)0, acc1, false, false);
        acc2 = __builtin_amdgcn_wmma_f32_16x16x4_f32(false, a, false, b2, (short)0, acc2, false, false);
        acc3 = __builtin_amdgcn_wmma_f32_16x16x4_f32(false, a, false, b3, (short)0, acc3, false, false);
    }

    float bA = 0.f, bB = 0.f, bC = 0.f, bD = 0.f;
    if (bias) { bA = bias[m]; bB = bias[m + 16]; bC = bias[m + 32]; bD = bias[m + 48]; }

    #pragma unroll
    for (int i = 0; i < 8; ++i) {
        int orow = rowBase + hi8 + i;
        if (orow < M) {
            float v0 = acc0[i] + bA, v1 = acc1[i] + bB;
            float v2 = acc2[i] + bC, v3 = acc3[i] + bD;
            if (applyRelu) {
                v0 = fmaxf(v0, 0.f); v1 = fmaxf(v1, 0.f);
                v2 = fmaxf(v2, 0.f); v3 = fmaxf(v3, 0.f);
            }
            float* drow = D + (size_t)orow * HID;
            drow[m]      = v0; drow[m + 16] = v1;
            drow[m + 32] = v2; drow[m + 48] = v3;
        }
    }
}

// ---------------------------------------------------------------------------
// Per-graph conditioning: layernorm(climber) -> c = relu(cn@W_c+b_c) in LDS,
// then gb1 = c@Wf1+bf1, gb2 = c@Wf2+bf2 (128 cols each). 1000 blocks x 128 thr.
// ---------------------------------------------------------------------------
__global__ __launch_bounds__(128) void gat_climber(
    const float* __restrict__ climber, const float* __restrict__ ln_g,
    const float* __restrict__ ln_b,    const float* __restrict__ W_c,
    const float* __restrict__ b_c,     const float* __restrict__ Wf1,
    const float* __restrict__ bf1,     const float* __restrict__ Wf2,
    const float* __restrict__ bf2,     float* __restrict__ gb1,
    float* __restrict__ gb2)
{
    __shared__ float cs[HID];
    const int g = blockIdx.x;
    const int t = threadIdx.x;

    float v[6]; float mu = 0.f;
    #pragma unroll
    for (int k = 0; k < 6; ++k) { v[k] = climber[g * 6 + k]; mu += v[k]; }
    mu *= (1.0f / 6.0f);
    float var = 0.f;
    #pragma unroll
    for (int k = 0; k < 6; ++k) { float d = v[k] - mu; var += d * d; }
    var *= (1.0f / 6.0f);
    const float rs = rsqrtf(var + 1e-5f);

    if (t < HID) {
        float acc = b_c[t];
        #pragma unroll
        for (int k = 0; k < 6; ++k) {
            float cn = (v[k] - mu) * rs * ln_g[k] + ln_b[k];
            acc += cn * W_c[k * HID + t];
        }
        cs[t] = fmaxf(acc, 0.f);
    }
    __syncthreads();

    float a1 = bf1[t], a2 = bf2[t];
    for (int k = 0; k < HID; ++k) {
        float cv = cs[k];
        a1 += cv * Wf1[k * 128 + t];
        a2 += cv * Wf2[k * 128 + t];
    }
    gb1[g * 128 + t] = a1;
    gb2[g * 128 + t] = a2;
}

// h = (x[:, :6] @ W_in + b_in) with FiLM1 applied. One thread per (node, hid).
__global__ void gat_node_init(
    const float* __restrict__ x, const int* __restrict__ batch,
    const float* __restrict__ W_in, const float* __restrict__ b_in,
    const float* __restrict__ gb1, float* __restrict__ h)
{
    int idx = blockIdx.x * blockDim.x + threadIdx.x;
    if (idx >= NN * HID) return;
    int n = idx >> 6, j = idx & 63;
    float acc = b_in[j];
    #pragma unroll
    for (int k = 0; k < 6; ++k) acc += x[n * 8 + k] * W_in[k * HID + j];
    int g = batch[n];
    float gamma = gb1[g * 128 + j];
    float beta  = gb1[g * 128 + 64 + j];
    h[idx] = acc * (1.0f + gamma) + beta;
}

// out = ALPHA_FLAG * ( relu(flags@Wh1+bh1) @ Wh2 + bh2 ). One thread per node.
__global__ void gat_flag_head(
    const float* __restrict__ x, const float* __restrict__ Wh1,
    const float* __restrict__ bh1, const float* __restrict__ Wh2,
    const float* __restrict__ bh2, float* __restrict__ out)
{
    int n = blockIdx.x * blockDim.x + threadIdx.x;
    if (n >= NN) return;
    float f0 = x[n * 8 + 6], f1 = x[n * 8 + 7];
    float p[8];
    #pragma unroll
    for (int j = 0; j < 8; ++j)
        p[j] = fmaxf(f0 * Wh1[j] + f1 * Wh1[8 + j] + bh1[j], 0.f);
    #pragma unroll
    for (int o = 0; o < OUTC; ++o) {
        float s = bh2[o];
        #pragma unroll
        for (int j = 0; j < 8; ++j) s += p[j] * Wh2[j * OUTC + o];
        out[n * OUTC + o] = ALPHA_FLAG * s;
    }
}

// zs[n] = z[n] . a_src ;  zd[n] = z[n] . a_dst   (b128 row loads)
__global__ void gat_scores(
    const float* __restrict__ z, const float* __restrict__ a_src,
    const float* __restrict__ a_dst, float* __restrict__ zs, float* __restrict__ zd)
{
    int n = blockIdx.x * blockDim.x + threadIdx.x;
    if (n >= NN) return;
    float ss = 0.f, sd = 0.f;
    const v4f* zr = (const v4f*)(z + (size_t)n * HID);
    #pragma unroll
    for (int q = 0; q < HID / 4; ++q) {
        v4f zv = zr[q];
        #pragma unroll
        for (int c = 0; c < 4; ++c) {
            int k = q * 4 + c;
            ss += zv[c] * a_src[k];
            sd += zv[c] * a_dst[k];
        }
    }
    zs[n] = ss; zd[n] = sd;
}

__global__ void gat_fill_neginf(float* __restrict__ m)
{
    int n = blockIdx.x * blockDim.x + threadIdx.x;
    if (n < NN) m[n] = -INFINITY;
}

__device__ __forceinline__ void atomicMaxFloat(float* addr, float value)
{
    if (value >= 0.f) atomicMax((int*)addr, __float_as_int(value));
    else              atomicMin((unsigned int*)addr, __float_as_uint(value));
}

__device__ __forceinline__ void edge_endpoints(const int* ei, int e, int& s, int& d)
{
    if (e < NE) { s = ei[e]; d = ei[NE + e]; }
    else        { s = d = e - NE; }
}

// pass 1: segment max of leaky_relu attention logits
__global__ void gat_edge_max(
    const int* __restrict__ ei, const float* __restrict__ zs,
    const float* __restrict__ zd, float* __restrict__ m)
{
    int e = blockIdx.x * blockDim.x + threadIdx.x;
    if (e >= ET) return;
    int s, d; edge_endpoints(ei, e, s, d);
    float el = zs[s] + zd[d];
    el = (el > 0.f) ? el : NEG_SLOPE * el;
    atomicMaxFloat(&m[d], el);
}

// pass 2: ex = exp(e - m[dst]) ; segment sum into ssum
__global__ void gat_edge_exp(
    const int* __restrict__ ei, const float* __restrict__ zs,
    const float* __restrict__ zd, const float* __restrict__ m,
    float* __restrict__ ex, float* __restrict__ ssum)
{
    int e = blockIdx.x * blockDim.x + threadIdx.x;
    if (e >= ET) return;
    int s, d; edge_endpoints(ei, e, s, d);
    float el = zs[s] + zd[d];
    el = (el > 0.f) ? el : NEG_SLOPE * el;
    float ev = expf(el - m[d]);
    ex[e] = ev;
    atomicAdd(&ssum[d], ev);
}

// pass 3: agg[dst] += alpha * z[src]  (b128 source reads, per-float atomics)
__global__ void gat_edge_scatter(
    const int* __restrict__ ei, const float* __restrict__ z,
    const float* __restrict__ ex, const float* __restrict__ ssum,
    float* __restrict__ agg)
{
    int e = blockIdx.x * blockDim.x + threadIdx.x;
    if (e >= ET) return;
    int s, d; edge_endpoints(ei, e, s, d);
    float alpha = ex[e] / (ssum[d] + 1e-16f);
    const v4f* zr = (const v4f*)(z + (size_t)s * HID);
    float* ar = agg + (size_t)d * HID;
    #pragma unroll
    for (int q = 0; q < HID / 4; ++q) {
        v4f zv = zr[q];
        #pragma unroll
        for (int c = 0; c < 4; ++c)
            atomicAdd(&ar[q * 4 + c], alpha * zv[c]);
    }
}

// h_out = relu(agg + bg) , optionally FiLM'd with gb. Thread per (node, hid).
__global__ void gat_finalize(
    const float* __restrict__ agg, const float* __restrict__ bg,
    const int* __restrict__ batch, const float* __restrict__ gb,
    float* __restrict__ hout)
{
    int idx = blockIdx.x * blockDim.x + threadIdx.x;
    if (idx >= NN * HID) return;
    int n = idx >> 6, j = idx & 63;
    float v = fmaxf(agg[idx] + bg[j], 0.f);
    if (gb) {
        int g = batch[n];
        v = v * (1.0f + gb[g * 128 + j]) + gb[g * 128 + 64 + j];
    }
    hout[idx] = v;
}

// out += t @ Wc2 + bc2  (64 -> 4). One thread per node, b128 row loads.
__global__ void gat_cls_out(
    const float* __restrict__ t, const float* __restrict__ Wc2,
    const float* __restrict__ bc2, float* __restrict__ out)
{
    int n = blockIdx.x * blockDim.x + threadIdx.x;
    if (n >= NN) return;
    float acc[OUTC];
    #pragma unroll
    for (int o = 0; o < OUTC; ++o) acc[o] = bc2[o];
    const v4f* tr = (const v4f*)(t + (size_t)n * HID);
    for (int q = 0; q < HID / 4; ++q) {
        v4f tv = tr[q];
        #pragma unroll
        for (int c = 0; c < 4; ++c) {
            const float* wrow = Wc2 + (q * 4 + c) * OUTC;
            #pragma unroll
            for (int o = 0; o < OUTC; ++o) acc[o] += tv[c] * wrow[o];
        }
    }
    #pragma unroll
    for (int o = 0; o < OUTC; ++o) out[n * OUTC + o] += acc[o];
}

// ---------------------------------------------------------------------------
extern "C" void kernel_launch(void* const* d_in, const int* in_sizes, int n_in,
                              void* d_out, int out_size, void* d_ws, size_t ws_size,
                              hipStream_t stream)
{
    const float* x        = (const float*)d_in[0];
    const int*   ei       = (const int*)  d_in[1];
    const int*   batch    = (const int*)  d_in[2];
    const float* climber  = (const float*)d_in[3];
    const float* W_in     = (const float*)d_in[4];
    const float* b_in     = (const float*)d_in[5];
    const float* ln_g     = (const float*)d_in[6];
    const float* ln_b     = (const float*)d_in[7];
    const float* W_c      = (const float*)d_in[8];
    const float* b_c      = (const float*)d_in[9];
    const float* Wf1      = (const float*)d_in[10];
    const float* bf1      = (const float*)d_in[11];
    const float* Wf2      = (const float*)d_in[12];
    const float* bf2      = (const float*)d_in[13];
    const float* Wg1      = (const float*)d_in[14];
    const float* as1      = (const float*)d_in[15];
    const float* ad1      = (const float*)d_in[16];
    const float* bg1      = (const float*)d_in[17];
    const float* Wg2      = (const float*)d_in[18];
    const float* as2      = (const float*)d_in[19];
    const float* ad2      = (const float*)d_in[20];
    const float* bg2      = (const float*)d_in[21];
    const float* Wc1      = (const float*)d_in[22];
    const float* bc1      = (const float*)d_in[23];
    const float* Wc2      = (const float*)d_in[24];
    const float* bc2      = (const float*)d_in[25];
    const float* Wh1      = (const float*)d_in[26];
    const float* bh1      = (const float*)d_in[27];
    const float* Wh2      = (const float*)d_in[28];
    const float* bh2      = (const float*)d_in[29];
    float* out = (float*)d_out;

    // workspace carve-up (floats)
    float* ws   = (float*)d_ws;
    float* h    = ws;                         // NN*64
    float* z    = h    + (size_t)NN * HID;    // NN*64
    float* agg  = z    + (size_t)NN * HID;    // NN*64
    float* zs   = agg  + (size_t)NN * HID;    // NN
    float* zd   = zs   + NN;                  // NN
    float* mbuf = zd   + NN;                  // NN
    float* sbuf = mbuf + NN;                  // NN
    float* ex   = sbuf + NN;                  // ET
    float* gb1  = ex   + ET;                  // NG*128
    float* gb2  = gb1  + NG * 128;            // NG*128

    const int B = 256;
    const int gNodeHid = (NN * HID + B - 1) / B;
    const int gNode    = (NN + B - 1) / B;
    const int gEdge    = (ET + B - 1) / B;
    const int gGemm    = (NN + 63) / 64;      // 64 rows per block (4 waves x 16)

    // conditioning
    gat_climber<<<NG, 128, 0, stream>>>(climber, ln_g, ln_b, W_c, b_c,
                                        Wf1, bf1, Wf2, bf2, gb1, gb2);
    // node embedding + FiLM1
    gat_node_init<<<gNodeHid, B, 0, stream>>>(x, batch, W_in, b_in, gb1, h);
    // flag head contribution into out
    gat_flag_head<<<gNode, B, 0, stream>>>(x, Wh1, bh1, Wh2, bh2, out);

    // ---- GAT layer 1 ----
    gat_gemm64_wmma<<<gGemm, 128, 0, stream>>>(h, Wg1, nullptr, z, NN, 0);
    gat_scores<<<gNode, B, 0, stream>>>(z, as1, ad1, zs, zd);
    gat_fill_neginf<<<gNode, B, 0, stream>>>(mbuf);
    hipMemsetAsync(sbuf, 0, (size_t)NN * sizeof(float), stream);
    hipMemsetAsync(agg,  0, (size_t)NN * HID * sizeof(float), stream);
    gat_edge_max<<<gEdge, B, 0, stream>>>(ei, zs, zd, mbuf);
    gat_edge_exp<<<gEdge, B, 0, stream>>>(ei, zs, zd, mbuf, ex, sbuf);
    gat_edge_scatter<<<gEdge, B, 0, stream>>>(ei, z, ex, sbuf, agg);
    gat_finalize<<<gNodeHid, B, 0, stream>>>(agg, bg1, batch, gb2, h);  // relu + FiLM2

    // ---- GAT layer 2 ----
    gat_gemm64_wmma<<<gGemm, 128, 0, stream>>>(h, Wg2, nullptr, z, NN, 0);
    gat_scores<<<gNode, B, 0, stream>>>(z, as2, ad2, zs, zd);
    gat_fill_neginf<<<gNode, B, 0, stream>>>(mbuf);
    hipMemsetAsync(sbuf, 0, (size_t)NN * sizeof(float), stream);
    hipMemsetAsync(agg,  0, (size_t)NN * HID * sizeof(float), stream);
    gat_edge_max<<<gEdge, B, 0, stream>>>(ei, zs, zd, mbuf);
    gat_edge_exp<<<gEdge, B, 0, stream>>>(ei, zs, zd, mbuf, ex, sbuf);
    gat_edge_scatter<<<gEdge, B, 0, stream>>>(ei, z, ex, sbuf, agg);
    gat_finalize<<<gNodeHid, B, 0, stream>>>(agg, bg2, batch, nullptr, h); // relu only

    // ---- classifier head ----
    gat_gemm64_wmma<<<gGemm, 128, 0, stream>>>(h, Wc1, bc1, z, NN, 1);   // t = relu(h@Wc1+bc1)
    gat_cls_out<<<gNode, B, 0, stream>>>(z, Wc2, bc2, out);              // out += t@Wc2+bc2
}